// OnetPlusPlus2_24077586661646
// MI455X (gfx1250) — compile-verified
//
#include <hip/hip_runtime.h>
#include <math.h>

#define B_    2
#define NQ    4096
#define NP    8192
#define D_    128
#define KNN   16
#define M_TOT (B_ * NQ)        // 8192 flattened queries
#define LEAKY 0.2f

typedef float v2f __attribute__((ext_vector_type(2)));
typedef float v8f __attribute__((ext_vector_type(8)));

// ---------------------------------------------------------------------------
// Kernel 1: KNN top-16 by squared L2.  One thread per query, point cloud
// streamed through LDS in 1024-point tiles (SoA to avoid bank conflicts).
// Sorted top-16 (ascending d2) lives in VGPRs; strict '<' keeps earliest
// index on ties, matching top_k tie-break order.
// ---------------------------------------------------------------------------
__global__ __launch_bounds__(256) void knn_kernel(const float* __restrict__ query,
                                                  const float* __restrict__ global_pc,
                                                  int* __restrict__ idx_out) {
    const int b     = blockIdx.x >> 4;           // 16 blocks per batch
    const int q     = ((blockIdx.x & 15) << 8) + threadIdx.x;
    const int m     = b * NQ + q;

    const float qx = query[m * 3 + 0];
    const float qy = query[m * 3 + 1];
    const float qz = query[m * 3 + 2];

    __shared__ float spx[1024];
    __shared__ float spy[1024];
    __shared__ float spz[1024];

    float bestd[KNN];
    int   besti[KNN];
#pragma unroll
    for (int k = 0; k < KNN; ++k) { bestd[k] = 3.4e38f; besti[k] = 0; }

    const float* __restrict__ pc = global_pc + (size_t)b * NP * 3;

    for (int tile = 0; tile < NP / 1024; ++tile) {
        const int base = tile * 1024;
        __syncthreads();
        for (int i = threadIdx.x; i < 1024; i += 256) {
            const float* p = pc + (size_t)(base + i) * 3;
            spx[i] = p[0]; spy[i] = p[1]; spz[i] = p[2];
        }
        // prefetch next tile while we crunch this one (global_prefetch_b8)
        if (tile + 1 < NP / 1024)
            __builtin_prefetch(pc + (size_t)(base + 1024 + threadIdx.x * 4) * 3, 0, 0);
        __syncthreads();

        for (int i = 0; i < 1024; ++i) {
            const float dx = qx - spx[i];
            const float dy = qy - spy[i];
            const float dz = qz - spz[i];
            const float d2 = dx * dx + dy * dy + dz * dz;
            if (d2 < bestd[KNN - 1]) {
                int pos = KNN - 1;
#pragma unroll
                for (int t = KNN - 2; t >= 0; --t)
                    if (d2 < bestd[t]) pos = t;
#pragma unroll
                for (int t = KNN - 1; t > 0; --t)
                    if (t > pos) { bestd[t] = bestd[t - 1]; besti[t] = besti[t - 1]; }
                bestd[pos] = d2;
                besti[pos] = base + i;
            }
        }
    }

#pragma unroll
    for (int k = 0; k < KNN; ++k) idx_out[(size_t)m * KNN + k] = besti[k];
}

// ---------------------------------------------------------------------------
// Kernel 2: fused gather + positional embedding + attention + softmax fold.
// One wave32 per query; lane owns 4 of the 128 feature dims.  Emits
// hbar[m][d] = sum_k softmax(scores)_k * h_k[d].   (The value projection is
// linear and softmax sums to 1, so W_v can be applied AFTER this contraction
// — exactly equivalent, 16x less GEMM work.)
// ---------------------------------------------------------------------------
__global__ __launch_bounds__(256) void attn_kernel(const float* __restrict__ query,
                                                   const float* __restrict__ global_pc,
                                                   const float* __restrict__ point_feature,
                                                   const float* __restrict__ w_pe,
                                                   const float* __restrict__ b_pe,
                                                   const float* __restrict__ w_attn,
                                                   const int* __restrict__ knn_idx,
                                                   float* __restrict__ hbar) {
    const int wave = threadIdx.x >> 5;
    const int lane = threadIdx.x & 31;
    const int m    = blockIdx.x * 8 + wave;      // flattened query id
    const int b    = m >> 12;
    const int d0   = lane * 4;

    float wx[4], wy[4], wz[4], bp[4], wa[4];
#pragma unroll
    for (int t = 0; t < 4; ++t) {
        wx[t] = w_pe[(d0 + t) * 3 + 0];
        wy[t] = w_pe[(d0 + t) * 3 + 1];
        wz[t] = w_pe[(d0 + t) * 3 + 2];
        bp[t] = b_pe[d0 + t];
        wa[t] = w_attn[d0 + t];
    }

    const float qx = query[m * 3 + 0];
    const float qy = query[m * 3 + 1];
    const float qz = query[m * 3 + 2];

    float hreg[KNN][4];
    float score[KNN];

#pragma unroll
    for (int k = 0; k < KNN; ++k) {
        const int j = knn_idx[(size_t)m * KNN + k];
        const float* pp = global_pc + ((size_t)b * NP + j) * 3;
        const float px = qx - pp[0];
        const float py = qy - pp[1];
        const float pz = qz - pp[2];
        const float* pf = point_feature + ((size_t)b * NP + j) * D_ + d0;
        float s = 0.f;
#pragma unroll
        for (int t = 0; t < 4; ++t) {
            const float hv = pf[t] + wx[t] * px + wy[t] * py + wz[t] * pz + bp[t];
            hreg[k][t] = hv;
            s += hv * wa[t];
        }
        // wave32 butterfly reduction -> every lane holds the full dot product
#pragma unroll
        for (int off = 16; off >= 1; off >>= 1)
            s += __shfl_xor(s, off, 32);
        score[k] = s;
    }

    float mx = score[0];
#pragma unroll
    for (int k = 1; k < KNN; ++k) mx = fmaxf(mx, score[k]);
    float sum = 0.f;
#pragma unroll
    for (int k = 0; k < KNN; ++k) { score[k] = __expf(score[k] - mx); sum += score[k]; }
    const float inv = 1.0f / sum;

    float acc[4] = {0.f, 0.f, 0.f, 0.f};
#pragma unroll
    for (int k = 0; k < KNN; ++k)
#pragma unroll
        for (int t = 0; t < 4; ++t) acc[t] += score[k] * hreg[k][t];

#pragma unroll
    for (int t = 0; t < 4; ++t)
        hbar[(size_t)m * D_ + d0 + t] = acc[t] * inv;
}

// ---------------------------------------------------------------------------
// Kernel 3: query_feature = hbar @ W_v^T + b_v via V_WMMA_F32_16X16X4_F32.
// Block = 8 waves sharing one 16-row A tile (staged in LDS); wave w owns
// output N-tile [16w, 16w+16).  K=128 -> 32 WMMA issues per wave.
// A layout (ISA 16x4 f32): lanes 0-15 carry K=k0/k0+1, lanes 16-31 K=k0+2/3.
// Stores transposed into out[b, n, q] (channels-first output).
// ---------------------------------------------------------------------------
__global__ __launch_bounds__(256) void gemm_v_kernel(const float* __restrict__ hbar,
                                                     const float* __restrict__ w_v,
                                                     const float* __restrict__ b_v,
                                                     float* __restrict__ out) {
    __shared__ float As[16 * D_];                // 8 KB A tile, shared by 8 waves
    const int tileM = blockIdx.x * 16;

    for (int i = threadIdx.x; i < 16 * D_; i += 256)
        As[i] = hbar[(size_t)(tileM + (i >> 7)) * D_ + (i & (D_ - 1))];
    __syncthreads();

    const int wave  = threadIdx.x >> 5;
    const int lane  = threadIdx.x & 31;
    const int tileN = wave * 16;
    const int half  = lane >> 4;                 // 0: K=k0..k0+1   1: K=k0+2..k0+3
    const int l     = lane & 15;

    v8f c = {};
    for (int k0 = 0; k0 < D_; k0 += 4) {
        const int k = k0 + half * 2;
        v2f a, bb;
        a.x  = As[l * D_ + k];
        a.y  = As[l * D_ + k + 1];
        bb.x = w_v[(size_t)(tileN + l) * D_ + k];       // B[k][n] = W_v[n][k]
        bb.y = w_v[(size_t)(tileN + l) * D_ + k + 1];
        c = __builtin_amdgcn_wmma_f32_16x16x4_f32(
                /*neg_a=*/false, a, /*neg_b=*/false, bb,
                /*c_mod=*/(short)0, c, /*reuse_a=*/false, /*reuse_b=*/false);
    }

    const float bias = b_v[tileN + l];
#pragma unroll
    for (int r = 0; r < 8; ++r) {
        const int mrow = tileM + r + half * 8;   // C/D layout: VGPR r -> M=r / r+8
        const int n    = tileN + l;
        const int b    = mrow >> 12;
        const int q    = mrow & (NQ - 1);
        out[(size_t)b * 256 * NQ + (size_t)n * NQ + q] = c[r] + bias;
    }
}

// ---------------------------------------------------------------------------
// Kernel 4: position_feature = LeakyReLU(query @ W_pos^T + b_pos), written to
// out[b, 128+d, q].  K=3 GEMV — pure VALU, negligible cost.
// ---------------------------------------------------------------------------
__global__ __launch_bounds__(256) void posfeat_kernel(const float* __restrict__ query,
                                                      const float* __restrict__ w_pos,
                                                      const float* __restrict__ b_pos,
                                                      float* __restrict__ out) {
    const int gid = blockIdx.x * 256 + threadIdx.x;   // over M_TOT * D_
    const int m   = gid >> 7;
    const int d   = gid & (D_ - 1);
    float v = w_pos[d * 3 + 0] * query[m * 3 + 0]
            + w_pos[d * 3 + 1] * query[m * 3 + 1]
            + w_pos[d * 3 + 2] * query[m * 3 + 2]
            + b_pos[d];
    v = (v >= 0.f) ? v : LEAKY * v;
    const int b = m >> 12;
    const int q = m & (NQ - 1);
    out[(size_t)b * 256 * NQ + (size_t)(D_ + d) * NQ + q] = v;
}

// ---------------------------------------------------------------------------
extern "C" void kernel_launch(void* const* d_in, const int* in_sizes, int n_in,
                              void* d_out, int out_size, void* d_ws, size_t ws_size,
                              hipStream_t stream) {
    const float* query         = (const float*)d_in[0];
    const float* global_pc     = (const float*)d_in[1];
    const float* point_feature = (const float*)d_in[2];
    const float* w_pos         = (const float*)d_in[3];
    const float* b_pos         = (const float*)d_in[4];
    const float* w_pe          = (const float*)d_in[5];
    const float* b_pe          = (const float*)d_in[6];
    const float* w_v           = (const float*)d_in[7];
    const float* b_v           = (const float*)d_in[8];
    const float* w_attn        = (const float*)d_in[9];
    float* out = (float*)d_out;

    // workspace layout: [idx: 8192*16 int][hbar: 8192*128 float]
    int*   idx_ws  = (int*)d_ws;
    float* hbar_ws = (float*)((char*)d_ws + (size_t)M_TOT * KNN * sizeof(int));

    // K1: KNN. 32 blocks (16 per batch) x 256 threads, 1 query/thread.
    knn_kernel<<<M_TOT / 256, 256, 0, stream>>>(query, global_pc, idx_ws);

    // K2: fused attention -> hbar. 8 queries (waves) per block.
    attn_kernel<<<M_TOT / 8, 256, 0, stream>>>(query, global_pc, point_feature,
                                               w_pe, b_pe, w_attn, idx_ws, hbar_ws);

    // K3: WMMA f32 GEMM (8192x128 @ 128x128) -> out channels [0,128).
    gemm_v_kernel<<<M_TOT / 16, 256, 0, stream>>>(hbar_ws, w_v, b_v, out);

    // K4: leaky position embedding -> out channels [128,256).
    posfeat_kernel<<<(M_TOT * D_) / 256, 256, 0, stream>>>(query, w_pos, b_pos, out);
}